// FragNetFineTune_22771916603960
// MI455X (gfx1250) — compile-verified
//
#include <hip/hip_runtime.h>

// FragNet GNN for MI455X (gfx1250, wave32).
// Memory/atomic-bound graph scatter + fp32 WMMA (v_wmma_f32_16x16x4_f32) GEMMs.

typedef float v2f __attribute__((ext_vector_type(2)));
typedef float v8f __attribute__((ext_vector_type(8)));

#define N_ATOMS  131072
#define E_ATOM   262144
#define E_BOND   262144
#define N_FRAGS  16384
#define E_FRAG   32768
#define N_GRAPHS 4096
#define EMB      128
#define ATOM_F   45
#define EDGE_F   12

// ---------------- utility ----------------
__global__ void k_zero(float* __restrict__ p, int n) {
  int i = blockIdx.x * blockDim.x + threadIdx.x;
  if (i < n) p[i] = 0.f;
}

// ---------------- bond graph ----------------
// edge_attr[row] = node_b[row] + (1.5*bW + bb)   (self-loop folded into init)
// node_b[row] = nfb[row,0:12] @ eW + eb
__global__ void k_bond_init(const float* __restrict__ nfb,
                            const float* __restrict__ eW,
                            const float* __restrict__ eb,
                            const float* __restrict__ bW,
                            const float* __restrict__ bb,
                            float* __restrict__ eattr) {
  int gid = blockIdx.x * blockDim.x + threadIdx.x;
  int row = gid >> 5;
  int lane = gid & 31;
  if (row >= E_ATOM) return;
  float nf[EDGE_F];
#pragma unroll
  for (int k = 0; k < EDGE_F; ++k) nf[k] = nfb[(size_t)row * EDGE_F + k];
  int c0 = lane * 4;
  float4 acc;
  float* ap = &acc.x;
#pragma unroll
  for (int j = 0; j < 4; ++j) {
    int c = c0 + j;
    float s = eb[c] + 1.5f * bW[c] + bb[c];
#pragma unroll
    for (int k = 0; k < EDGE_F; ++k) s = fmaf(nf[k], eW[k * EMB + c], s);
    ap[j] = s;
  }
  *(float4*)(eattr + (size_t)row * EMB + c0) = acc;
}

// For each real bond-graph edge: eattr[tgt] += (eab*bW + bb) + (nfb[src] @ eW + eb)
// (recompute the 12->128 embed on the fly: 48B gather instead of 512B)
__global__ void k_bond_scatter(const int* __restrict__ eib,
                               const float* __restrict__ eab,
                               const float* __restrict__ nfb,
                               const float* __restrict__ eW,
                               const float* __restrict__ eb,
                               const float* __restrict__ bW,
                               const float* __restrict__ bb,
                               float* __restrict__ eattr) {
  int gid = blockIdx.x * blockDim.x + threadIdx.x;
  int e = gid >> 5;
  int lane = gid & 31;
  if (e >= E_BOND) return;
  int t = eib[e];            // edge_index_bonds_graph[0] = target
  int s = eib[E_BOND + e];   // edge_index_bonds_graph[1] = source
  float ea = eab[e];
  float nf[EDGE_F];
#pragma unroll
  for (int k = 0; k < EDGE_F; ++k) nf[k] = nfb[(size_t)s * EDGE_F + k];
  int c0 = lane * 4;
#pragma unroll
  for (int j = 0; j < 4; ++j) {
    int c = c0 + j;
    float v = fmaf(ea, bW[c], bb[c]) + eb[c];
#pragma unroll
    for (int k = 0; k < EDGE_F; ++k) v = fmaf(nf[k], eW[k * EMB + c], v);
    atomicAdd(&eattr[(size_t)t * EMB + c], v);
  }
}

// ---------------- fp32 WMMA GEMM ----------------
// out = [RELU_OUT](relu?[RELU_IN](A) @ W + bias),  A:[M,K] row-major, W:[K,N] row-major.
// One wave owns a 16x128 output tile (8 v8f accumulators); each A fragment
// feeds 8 back-to-back v_wmma_f32_16x16x4_f32. N=256 runs two 128-col passes.
// Compile-time K/N -> no bounds checks, immediate-offset loads, no 64-bit muls
// in the hot loop. M must be a multiple of 16.
template <int K, int N, bool RELU_IN, bool RELU_OUT, bool DUP>
__global__ void k_gemm_wmma(const float* __restrict__ A,
                            const float* __restrict__ W,
                            const float* __restrict__ bias,
                            float* __restrict__ out0,
                            float* __restrict__ out1,
                            int M) {
  int wave = (blockIdx.x * blockDim.x + threadIdx.x) >> 5;
  int lane = threadIdx.x & 31;
  int m0 = wave * 16;
  if (m0 >= M) return;                 // whole-wave uniform: EXEC stays all-ones
  int l = lane & 15;
  int hi = lane >> 4;                  // 0: k0,k0+1   1: k0+2,k0+3
  const float* Arow = A + (size_t)(m0 + l) * K + hi * 2;
  constexpr int KFULL = K / 4;

  for (int nc = 0; nc < N; nc += 128) {
    v8f acc[8];
#pragma unroll
    for (int nt = 0; nt < 8; ++nt) {
      float bv = bias[nc + nt * 16 + l];
      acc[nt][0] = bv; acc[nt][1] = bv; acc[nt][2] = bv; acc[nt][3] = bv;
      acc[nt][4] = bv; acc[nt][5] = bv; acc[nt][6] = bv; acc[nt][7] = bv;
    }
    const float* aP = Arow;
    const float* bP = W + (size_t)(hi * 2) * N + nc + l;
#pragma unroll 2
    for (int ks = 0; ks < KFULL; ++ks) {
      float a0 = aP[0], a1 = aP[1];
      if (RELU_IN) { a0 = fmaxf(a0, 0.f); a1 = fmaxf(a1, 0.f); }
      v2f a; a[0] = a0; a[1] = a1;
#pragma unroll
      for (int nt = 0; nt < 8; ++nt) {
        v2f b; b[0] = bP[nt * 16]; b[1] = bP[N + nt * 16];
        acc[nt] = __builtin_amdgcn_wmma_f32_16x16x4_f32(false, a, false, b,
                                                        (short)0, acc[nt],
                                                        false, false);
      }
      aP += 4;
      bP += 4 * N;
    }
    if (K % 4) {                       // ragged tail (K=45 only)
      int kb = KFULL * 4 + hi * 2;     // lane-varying
      float a0 = (kb < K) ? aP[0] : 0.f;
      float a1 = (kb + 1 < K) ? aP[1] : 0.f;
      if (RELU_IN) { a0 = fmaxf(a0, 0.f); a1 = fmaxf(a1, 0.f); }
      v2f a; a[0] = a0; a[1] = a1;
#pragma unroll
      for (int nt = 0; nt < 8; ++nt) {
        v2f b;
        b[0] = (kb < K) ? bP[nt * 16] : 0.f;
        b[1] = (kb + 1 < K) ? bP[N + nt * 16] : 0.f;
        acc[nt] = __builtin_amdgcn_wmma_f32_16x16x4_f32(false, a, false, b,
                                                        (short)0, acc[nt],
                                                        false, false);
      }
    }
    // D layout: VGPR v -> row m0+v (lanes 0-15) / m0+8+v (lanes 16-31), col = l
    float* o0 = out0 + (size_t)(m0 + hi * 8) * N + nc + l;
    float* o1 = DUP ? (out1 + (size_t)(m0 + hi * 8) * N + nc + l) : nullptr;
#pragma unroll
    for (int nt = 0; nt < 8; ++nt) {
#pragma unroll
      for (int v = 0; v < 8; ++v) {
        float x = acc[nt][v];
        if (RELU_OUT) x = fmaxf(x, 0.f);
        o0[(size_t)v * N + nt * 16] = x;
        if (DUP) o1[(size_t)v * N + nt * 16] = x;
      }
    }
  }
}

// ---------------- atom graph scatter ----------------
// acc[tgt] += eattr[e] + emb[src]   (self loop already in acc via dup-write)
__global__ void k_atom_scatter(const int* __restrict__ ei,
                               const float* __restrict__ eattr,
                               const float* __restrict__ emb,
                               float* __restrict__ acc) {
  int gid = blockIdx.x * blockDim.x + threadIdx.x;
  int e = gid >> 5;
  int lane = gid & 31;
  if (e >= E_ATOM) return;
  int s = ei[e];             // edge_index[0] = source
  int t = ei[E_ATOM + e];    // edge_index[1] = target
  int c0 = lane * 4;
  float4 ea = *(const float4*)(eattr + (size_t)e * EMB + c0);
  float4 xs = *(const float4*)(emb + (size_t)s * EMB + c0);
  float* dst = acc + (size_t)t * EMB + c0;
  atomicAdd(dst + 0, ea.x + xs.x);
  atomicAdd(dst + 1, ea.y + xs.y);
  atomicAdd(dst + 2, ea.z + xs.z);
  atomicAdd(dst + 3, ea.w + xs.w);
}

// ---------------- fragment / graph pooling ----------------
__global__ void k_seg_add128(const int* __restrict__ seg, const float* __restrict__ src,
                             float* __restrict__ dst, int nrows) {
  int gid = blockIdx.x * blockDim.x + threadIdx.x;
  int r = gid >> 5;
  int lane = gid & 31;
  if (r >= nrows) return;
  int g = seg[r];
  int c0 = lane * 4;
  float4 v = *(const float4*)(src + (size_t)r * EMB + c0);
  float* d = dst + (size_t)g * EMB + c0;
  atomicAdd(d + 0, v.x); atomicAdd(d + 1, v.y);
  atomicAdd(d + 2, v.z); atomicAdd(d + 3, v.w);
}

// pooled row stride 256: atoms -> cols [0,128), frags -> cols [128,256); relu on load
__global__ void k_pool256(const int* __restrict__ seg, const float* __restrict__ src,
                          float* __restrict__ pooled, int nrows, int col_off) {
  int gid = blockIdx.x * blockDim.x + threadIdx.x;
  int r = gid >> 5;
  int lane = gid & 31;
  if (r >= nrows) return;
  int g = seg[r];
  int c0 = lane * 4;
  float4 v = *(const float4*)(src + (size_t)r * EMB + c0);
  v.x = fmaxf(v.x, 0.f); v.y = fmaxf(v.y, 0.f);
  v.z = fmaxf(v.z, 0.f); v.w = fmaxf(v.w, 0.f);
  float* d = pooled + (size_t)g * 256 + col_off + c0;
  atomicAdd(d + 0, v.x); atomicAdd(d + 1, v.y);
  atomicAdd(d + 2, v.z); atomicAdd(d + 3, v.w);
}

// frag-graph edge scatter: frag_sum[dst] += x_frags[src]
__global__ void k_frag_edge(const int* __restrict__ fi, const float* __restrict__ xfrag,
                            float* __restrict__ fsum) {
  int gid = blockIdx.x * blockDim.x + threadIdx.x;
  int e = gid >> 5;
  int lane = gid & 31;
  if (e >= E_FRAG) return;
  int s = fi[e];
  int t = fi[E_FRAG + e];
  int c0 = lane * 4;
  float4 v = *(const float4*)(xfrag + (size_t)s * EMB + c0);
  float* d = fsum + (size_t)t * EMB + c0;
  atomicAdd(d + 0, v.x); atomicAdd(d + 1, v.y);
  atomicAdd(d + 2, v.z); atomicAdd(d + 3, v.w);
}

// out[g] = dot(h[g,0:256], outW) + outb   (wave per graph, shuffle reduce)
__global__ void k_head(const float* __restrict__ h, const float* __restrict__ outW,
                       const float* __restrict__ outb, float* __restrict__ out) {
  int g = (blockIdx.x * blockDim.x + threadIdx.x) >> 5;
  int lane = threadIdx.x & 31;
  if (g >= N_GRAPHS) return;
  float s = 0.f;
#pragma unroll
  for (int i = 0; i < 8; ++i) {
    int c = lane + i * 32;
    s = fmaf(h[(size_t)g * 256 + c], outW[c], s);
  }
  for (int m = 16; m > 0; m >>= 1) s += __shfl_xor(s, m, 32);
  if (lane == 0) out[g] = s + outb[0];
}

// ---------------- driver ----------------
extern "C" void kernel_launch(void* const* d_in, const int* in_sizes, int n_in,
                              void* d_out, int out_size, void* d_ws, size_t ws_size,
                              hipStream_t stream) {
  (void)in_sizes; (void)n_in; (void)out_size; (void)ws_size;
  const float* x_atoms    = (const float*)d_in[0];
  const int*   edge_index = (const int*)d_in[1];
  const int*   frag_index = (const int*)d_in[2];
  const int*   a2f        = (const int*)d_in[3];
  const float* nfb        = (const float*)d_in[4];
  const int*   eib        = (const int*)d_in[5];
  const float* eab        = (const float*)d_in[6];
  const int*   batch      = (const int*)d_in[7];
  const int*   frag_batch = (const int*)d_in[8];
  // params pytree flattened in jax dict-key sorted order:
  // per layer l (base 9+l*10): aW, ab, bW, bb, eW, eb, f1W, f1b, f2W, f2b
  // then lin1W(49), lin1b(50), outW(51), outb(52)
  auto P = [&](int i) { return (const float*)d_in[i]; };
  const float* lin1W = P(49); const float* lin1b = P(50);
  const float* outW  = P(51); const float* outb  = P(52);

  char* ws = (char*)d_ws;
  size_t off = 0;
  auto alloc = [&](size_t bytes) {
    float* p = (float*)(ws + off);
    off += (bytes + 255) & ~(size_t)255;
    return p;
  };
  float* X0      = alloc((size_t)N_ATOMS * EMB * 4);
  float* X1      = alloc((size_t)N_ATOMS * EMB * 4);
  float* EMBB    = alloc((size_t)N_ATOMS * EMB * 4);
  float* EATTR   = alloc((size_t)E_ATOM * EMB * 4);
  float* XFRAG   = alloc((size_t)N_FRAGS * EMB * 4);
  float* FSUM    = alloc((size_t)N_FRAGS * EMB * 4);
  float* H1      = alloc((size_t)N_FRAGS * 256 * 4);
  float* XF      = alloc((size_t)N_FRAGS * EMB * 4);
  float* POOLED  = alloc((size_t)N_GRAPHS * 256 * 4);
  float* HH      = alloc((size_t)N_GRAPHS * 256 * 4);

  const int B = 256;
  auto nb = [](long long threads) { return (unsigned)((threads + 255) / 256); };

  const float* xin = x_atoms;
  float* Xcur = nullptr;
  for (int l = 0; l < 4; ++l) {
    const float* aW = P(9 + l * 10 + 0); const float* ab = P(9 + l * 10 + 1);
    const float* bW = P(9 + l * 10 + 2); const float* bb = P(9 + l * 10 + 3);
    const float* eW = P(9 + l * 10 + 4); const float* eb = P(9 + l * 10 + 5);
    float* Xout = (l & 1) ? X1 : X0;

    // bond graph: init (node embed + self loop), then scatter real edges
    k_bond_init<<<nb((long long)E_ATOM * 32), B, 0, stream>>>(nfb, eW, eb, bW, bb, EATTR);
    k_bond_scatter<<<nb((long long)E_BOND * 32), B, 0, stream>>>(eib, eab, nfb, eW, eb, bW, bb, EATTR);
    // atom embed (WMMA); duplicate write into Xout covers atom self-loops
    unsigned gw = nb((long long)(N_ATOMS / 16) * 32);
    if (l == 0)
      k_gemm_wmma<ATOM_F, EMB, false, false, true><<<gw, B, 0, stream>>>(
          xin, aW, ab, EMBB, Xout, N_ATOMS);
    else
      k_gemm_wmma<EMB, EMB, true, false, true><<<gw, B, 0, stream>>>(
          xin, aW, ab, EMBB, Xout, N_ATOMS);
    // atom scatter
    k_atom_scatter<<<nb((long long)E_ATOM * 32), B, 0, stream>>>(edge_index, EATTR, EMBB, Xout);

    xin = Xout; Xcur = Xout;
  }

  // fragment path — only the last layer's xf is ever used by the reference
  const float* f1W = P(9 + 3 * 10 + 6); const float* f1b = P(9 + 3 * 10 + 7);
  const float* f2W = P(9 + 3 * 10 + 8); const float* f2b = P(9 + 3 * 10 + 9);
  k_zero<<<nb(N_FRAGS * EMB), B, 0, stream>>>(XFRAG, N_FRAGS * EMB);
  k_zero<<<nb(N_FRAGS * EMB), B, 0, stream>>>(FSUM, N_FRAGS * EMB);
  k_zero<<<nb(N_GRAPHS * 256), B, 0, stream>>>(POOLED, N_GRAPHS * 256);
  // x_frags = segsum over atoms of PRE-relu x_atoms_new
  k_seg_add128<<<nb((long long)N_ATOMS * 32), B, 0, stream>>>(a2f, Xcur, XFRAG, N_ATOMS);
  k_frag_edge<<<nb((long long)E_FRAG * 32), B, 0, stream>>>(frag_index, XFRAG, FSUM);
  // frag MLP: relu(FSUM@f1W+f1b) @ f2W + f2b  (WMMA)
  k_gemm_wmma<EMB, 256, false, true, false>
      <<<nb((long long)(N_FRAGS / 16) * 32), B, 0, stream>>>(FSUM, f1W, f1b, H1, nullptr, N_FRAGS);
  k_gemm_wmma<256, EMB, false, false, false>
      <<<nb((long long)(N_FRAGS / 16) * 32), B, 0, stream>>>(H1, f2W, f2b, XF, nullptr, N_FRAGS);

  // graph pooling (relu applied on load); pooled = [atoms | frags] per graph
  k_pool256<<<nb((long long)N_ATOMS * 32), B, 0, stream>>>(batch, Xcur, POOLED, N_ATOMS, 0);
  k_pool256<<<nb((long long)N_FRAGS * 32), B, 0, stream>>>(frag_batch, XF, POOLED, N_FRAGS, 128);

  // head: h = relu(pooled @ lin1W + lin1b) (WMMA), then out = h @ outW + outb
  k_gemm_wmma<256, 256, false, true, false>
      <<<nb((long long)(N_GRAPHS / 16) * 32), B, 0, stream>>>(POOLED, lin1W, lin1b, HH, nullptr, N_GRAPHS);
  k_head<<<nb((long long)N_GRAPHS * 32), B, 0, stream>>>(HH, outW, outb, (float*)d_out);
}